// GatedDirGCNConv_71777493451332
// MI455X (gfx1250) — compile-verified
//
#include <hip/hip_runtime.h>
#include <math.h>

#define DD 128
#define PITCH 132   // LDS row pitch in floats (avoid 64-bank conflicts at stride 128)

typedef float v2f __attribute__((ext_vector_type(2)));
typedef float v8f __attribute__((ext_vector_type(8)));

// ---------------------------------------------------------------------------
// One 16x16 f32 output tile via V_WMMA_F32_16X16X4_F32, K = 128.
// A tile staged in LDS as [16][PITCH]; W is a 128x128 row-major global matrix.
// Fragment layouts per CDNA5 ISA 7.12.2:
//   A 16x4 : row = lane%16, K = k0 + 2*(lane/16) + j   (j = vgpr 0/1)
//   B 4x16 : col = lane%16, K = k0 + 2*(lane/16) + j
//   C/D    : row = i + 8*(lane/16), col = lane%16      (i = vgpr 0..7)
// ---------------------------------------------------------------------------
__device__ __forceinline__ v8f wmma_f32_tile(const float* sA,
                                             const float* __restrict__ W,
                                             int colbase, v8f acc) {
  const int lane = threadIdx.x & 31;
  const int half = lane >> 4;
  const int l    = lane & 15;
#pragma unroll
  for (int k0 = 0; k0 < DD; k0 += 4) {
    const int ka = k0 + 2 * half;
    v2f a, b;
    a.x = sA[l * PITCH + ka];
    a.y = sA[l * PITCH + ka + 1];
    b.x = W[(size_t)ka * DD + colbase + l];
    b.y = W[(size_t)(ka + 1) * DD + colbase + l];
    acc = __builtin_amdgcn_wmma_f32_16x16x4_f32(false, a, false, b,
                                                (short)0, acc, false, false);
  }
  return acc;
}

// ---------------------------------------------------------------------------
// Stage a full 16x128 tile into LDS via async global->LDS B128 copies
// (GLOBAL_LOAD_ASYNC_TO_LDS_B128, tracked by ASYNCcnt; bypasses VGPRs).
// Each lane moves 16 bytes; 256 threads x 2 passes = 8 KB tile.
// Every wave drains its own ASYNCcnt before the block barrier.
// ---------------------------------------------------------------------------
__device__ __forceinline__ void stage_tile_async(float* sA,
                                                 const float* __restrict__ src,
                                                 int rowbase) {
  const unsigned lds_base = (unsigned)(unsigned long long)(uintptr_t)sA;
  for (int i = threadIdx.x; i < 16 * (DD / 4); i += 256) {  // 512 float4 chunks
    const int r = i >> 5, c4 = i & 31;                      // 32 float4 per row
    const unsigned lds_addr = lds_base + (unsigned)((r * PITCH + c4 * 4) * 4);
    const float* g = src + (size_t)(rowbase + r) * DD + c4 * 4;
    asm volatile("global_load_async_to_lds_b128 %0, %1, off"
                 :: "v"(lds_addr), "v"(g)
                 : "memory");
  }
  asm volatile("s_wait_asynccnt 0x0" ::: "memory");
}

// Cooperatively stage a 16x128 row tile into LDS (blockDim.x == 256).
__device__ __forceinline__ void stage_tile(float* sA, const float* __restrict__ src,
                                           int rowbase, int nrows) {
  if (rowbase + 16 <= nrows) {
    stage_tile_async(sA, src, rowbase);          // full tile: async DMA path
  } else {
    for (int i = threadIdx.x; i < 16 * DD; i += 256) {
      int r = i >> 7, c = i & (DD - 1);
      int gr = rowbase + r;
      sA[r * PITCH + c] = (gr < nrows) ? src[(size_t)gr * DD + c] : 0.f;
    }
  }
}

// ---------------------------------------------------------------------------
// Kernel 0: zero workspace accumulators
// ---------------------------------------------------------------------------
__global__ void zero_kernel(float* __restrict__ p, size_t n) {
  size_t i = (size_t)blockIdx.x * blockDim.x + threadIdx.x;
  if (i < n) p[i] = 0.f;
}

// ---------------------------------------------------------------------------
// Kernel 1: A = x @ W_l1[:128],  B = x @ W_l1[128:] + b_l1
// ---------------------------------------------------------------------------
__global__ void gemm_lcs_pre(const float* __restrict__ x,
                             const float* __restrict__ W_l1,
                             const float* __restrict__ b_l1,
                             float* __restrict__ A, float* __restrict__ B, int N) {
  __shared__ float sX[16 * PITCH];
  const int rowbase = blockIdx.x * 16;
  stage_tile(sX, x, rowbase, N);
  __syncthreads();

  const int wave = threadIdx.x >> 5;
  const int colbase = wave * 16;
  const int lane = threadIdx.x & 31;
  const int half = lane >> 4, l = lane & 15;

  v8f accA = {}; v8f accB = {};
  accA = wmma_f32_tile(sX, W_l1, colbase, accA);
  accB = wmma_f32_tile(sX, W_l1 + (size_t)DD * DD, colbase, accB);
  const float bias = b_l1[colbase + l];
#pragma unroll
  for (int i = 0; i < 8; ++i) {
    int r = rowbase + i + 8 * half;
    if (r < N) {
      A[(size_t)r * DD + colbase + l] = accA[i];
      B[(size_t)r * DD + colbase + l] = accB[i] + bias;
    }
  }
}

// ---------------------------------------------------------------------------
// Kernel 2: one wave per edge.
//   h = relu(A[src] + B[dst]); lcs = sigmoid(h . W_l2 + b_l2); w = lcs*cnt
//   acc_in[dst]  += w * x[src];  acc_out[src] += w * x[dst]
//   t_in[dst]+=w; t_out[src]+=w; deg_in[dst]+=cnt; deg_out[src]+=cnt
// ---------------------------------------------------------------------------
__global__ void edge_kernel(const float* __restrict__ x,
                            const int* __restrict__ src, const int* __restrict__ dst,
                            const float* __restrict__ counts,
                            const float* __restrict__ A, const float* __restrict__ B,
                            const float* __restrict__ W_l2, const float* __restrict__ b_l2,
                            float* __restrict__ acc_in, float* __restrict__ acc_out,
                            float* __restrict__ t_in, float* __restrict__ t_out,
                            float* __restrict__ deg_in, float* __restrict__ deg_out,
                            int E) {
  const int gw = (int)((blockIdx.x * (size_t)blockDim.x + threadIdx.x) >> 5);
  const int lane = threadIdx.x & 31;
  if (gw >= E) return;

  const int s = src[gw];
  const int d = dst[gw];
  const float c = counts[gw];

  const float4 a = ((const float4*)(A + (size_t)s * DD))[lane];
  const float4 b = ((const float4*)(B + (size_t)d * DD))[lane];
  const float4 w2 = ((const float4*)W_l2)[lane];

  float p = fmaxf(a.x + b.x, 0.f) * w2.x +
            fmaxf(a.y + b.y, 0.f) * w2.y +
            fmaxf(a.z + b.z, 0.f) * w2.z +
            fmaxf(a.w + b.w, 0.f) * w2.w;
#pragma unroll
  for (int off = 16; off > 0; off >>= 1) p += __shfl_xor(p, off, 32);

  const float z = p + b_l2[0];
  const float lcs = 1.f / (1.f + __expf(-z));
  const float w = lcs * c;

  const float4 xs = ((const float4*)(x + (size_t)s * DD))[lane];
  const float4 xd = ((const float4*)(x + (size_t)d * DD))[lane];

  float* pin = acc_in + (size_t)d * DD + lane * 4;
  atomicAdd(pin + 0, w * xs.x);
  atomicAdd(pin + 1, w * xs.y);
  atomicAdd(pin + 2, w * xs.z);
  atomicAdd(pin + 3, w * xs.w);

  float* pout = acc_out + (size_t)s * DD + lane * 4;
  atomicAdd(pout + 0, w * xd.x);
  atomicAdd(pout + 1, w * xd.y);
  atomicAdd(pout + 2, w * xd.z);
  atomicAdd(pout + 3, w * xd.w);

  if (lane == 0) {
    atomicAdd(t_in + d, w);
    atomicAdd(t_out + s, w);
    atomicAdd(deg_in + d, c);
    atomicAdd(deg_out + s, c);
  }
}

// ---------------------------------------------------------------------------
// Kernel 3: h_in  = (acc_in @ W_sd + t_in*b_sd) / max(deg_in,1)
//           h_out = (acc_out@ W_ds + t_out*b_ds) / max(deg_out,1)
// ---------------------------------------------------------------------------
__global__ void gemm_msg_finalize(const float* __restrict__ acc_in,
                                  const float* __restrict__ acc_out,
                                  const float* __restrict__ W_sd, const float* __restrict__ b_sd,
                                  const float* __restrict__ W_ds, const float* __restrict__ b_ds,
                                  const float* __restrict__ t_in, const float* __restrict__ t_out,
                                  const float* __restrict__ deg_in, const float* __restrict__ deg_out,
                                  float* __restrict__ h_in, float* __restrict__ h_out, int N) {
  __shared__ float sIn[16 * PITCH];
  __shared__ float sOut[16 * PITCH];
  const int rowbase = blockIdx.x * 16;
  stage_tile(sIn, acc_in, rowbase, N);
  stage_tile(sOut, acc_out, rowbase, N);
  __syncthreads();

  const int wave = threadIdx.x >> 5;
  const int colbase = wave * 16;
  const int lane = threadIdx.x & 31;
  const int half = lane >> 4, l = lane & 15;

  v8f aIn = {}; v8f aOut = {};
  aIn  = wmma_f32_tile(sIn,  W_sd, colbase, aIn);
  aOut = wmma_f32_tile(sOut, W_ds, colbase, aOut);
  const float bin  = b_sd[colbase + l];
  const float bout = b_ds[colbase + l];
#pragma unroll
  for (int i = 0; i < 8; ++i) {
    int r = rowbase + i + 8 * half;
    if (r < N) {
      float di = fmaxf(deg_in[r], 1.f);
      float dq = fmaxf(deg_out[r], 1.f);
      h_in[(size_t)r * DD + colbase + l]  = (aIn[i]  + t_in[r]  * bin)  / di;
      h_out[(size_t)r * DD + colbase + l] = (aOut[i] + t_out[r] * bout) / dq;
    }
  }
}

// ---------------------------------------------------------------------------
// Kernel 4: G = relu(h_in@Wg1_top + h_out@Wg1_bot + b_g1)
//           gate = sigmoid(G @ W_g2 + b_g2)
//           out = gate*h_in + (1-gate)*h_out + x
// ---------------------------------------------------------------------------
__global__ void gate_fuse(const float* __restrict__ h_in, const float* __restrict__ h_out,
                          const float* __restrict__ x,
                          const float* __restrict__ W_g1, const float* __restrict__ b_g1,
                          const float* __restrict__ W_g2, const float* __restrict__ b_g2,
                          float* __restrict__ out, int N) {
  __shared__ float sIn[16 * PITCH];
  __shared__ float sOut[16 * PITCH];
  __shared__ float sG[16 * PITCH];
  __shared__ float sGate[16];
  const int rowbase = blockIdx.x * 16;
  stage_tile(sIn, h_in, rowbase, N);
  stage_tile(sOut, h_out, rowbase, N);
  __syncthreads();

  const int wave = threadIdx.x >> 5;
  const int colbase = wave * 16;
  const int lane = threadIdx.x & 31;
  const int half = lane >> 4, l = lane & 15;

  v8f acc = {};
  acc = wmma_f32_tile(sIn,  W_g1, colbase, acc);
  acc = wmma_f32_tile(sOut, W_g1 + (size_t)DD * DD, colbase, acc);
  const float bias = b_g1[colbase + l];
#pragma unroll
  for (int i = 0; i < 8; ++i) {
    int r = i + 8 * half;
    sG[r * PITCH + colbase + l] = fmaxf(acc[i] + bias, 0.f);
  }
  __syncthreads();

  if (threadIdx.x < 16) {
    float sum = b_g2[0];
    for (int k = 0; k < DD; ++k) sum += sG[threadIdx.x * PITCH + k] * W_g2[k];
    sGate[threadIdx.x] = 1.f / (1.f + __expf(-sum));
  }
  __syncthreads();

  for (int i = threadIdx.x; i < 16 * DD; i += 256) {
    int r = i >> 7, c = i & (DD - 1);
    int gr = rowbase + r;
    if (gr < N) {
      size_t idx = (size_t)gr * DD + c;
      float g = sGate[r];
      out[idx] = g * h_in[idx] + (1.f - g) * h_out[idx] + x[idx];
    }
  }
}

// ---------------------------------------------------------------------------
extern "C" void kernel_launch(void* const* d_in, const int* in_sizes, int n_in,
                              void* d_out, int out_size, void* d_ws, size_t ws_size,
                              hipStream_t stream) {
  const float* x      = (const float*)d_in[0];
  const int*   src    = (const int*)  d_in[1];
  const int*   dst    = (const int*)  d_in[2];
  const float* counts = (const float*)d_in[3];
  const float* W_sd   = (const float*)d_in[4];
  const float* b_sd   = (const float*)d_in[5];
  const float* W_ds   = (const float*)d_in[6];
  const float* b_ds   = (const float*)d_in[7];
  const float* W_l1   = (const float*)d_in[8];
  const float* b_l1   = (const float*)d_in[9];
  const float* W_l2   = (const float*)d_in[10];
  const float* b_l2   = (const float*)d_in[11];
  const float* W_g1   = (const float*)d_in[12];
  const float* b_g1   = (const float*)d_in[13];
  const float* W_g2   = (const float*)d_in[14];
  const float* b_g2   = (const float*)d_in[15];

  const int N = in_sizes[0] / DD;
  const int E = in_sizes[1];

  float* ws = (float*)d_ws;
  const size_t nd = (size_t)N * DD;
  float* A      = ws;                 // -> reused as h_in
  float* B      = ws + nd;            // -> reused as h_out
  float* accI   = ws + 2 * nd;
  float* accO   = ws + 3 * nd;
  float* t_in   = ws + 4 * nd;
  float* t_out  = t_in + N;
  float* deg_in = t_out + N;
  float* deg_out= deg_in + N;
  float* hI = A;
  float* hO = B;

  const size_t zn = 2 * nd + 4 * (size_t)N;   // accI..deg_out contiguous
  zero_kernel<<<(unsigned)((zn + 255) / 256), 256, 0, stream>>>(accI, zn);

  const int tiles = (N + 15) / 16;
  gemm_lcs_pre<<<tiles, 256, 0, stream>>>(x, W_l1, b_l1, A, B, N);

  const int eblocks = (E + 7) / 8;            // 8 waves per 256-thread block
  edge_kernel<<<eblocks, 256, 0, stream>>>(x, src, dst, counts, A, B, W_l2, b_l2,
                                           accI, accO, t_in, t_out, deg_in, deg_out, E);

  gemm_msg_finalize<<<tiles, 256, 0, stream>>>(accI, accO, W_sd, b_sd, W_ds, b_ds,
                                               t_in, t_out, deg_in, deg_out, hI, hO, N);

  gate_fuse<<<tiles, 256, 0, stream>>>(hI, hO, x, W_g1, b_g1, W_g2, b_g2,
                                       (float*)d_out, N);
}